// AssemblyAwareListMLELoss_30666066493662
// MI455X (gfx1250) — compile-verified
//
#include <hip/hip_runtime.h>
#include <math.h>

// ListMLE loss: B=4096 rows, N=8192 logits/row, L=256 positives/row.
// One workgroup (256 threads = 8 wave32) per row. Suffix-sum of exp values
// computed as a 16x16x16 f32 matmul against a triangular ones matrix using
// V_WMMA_F32_16X16X4_F32 (4 chained K=4 steps) on wave 0.

#define BB   4096
#define NN   8192
#define LL   256
#define EPSV 1e-8f

typedef float v2f __attribute__((ext_vector_type(2)));
typedef float v8f __attribute__((ext_vector_type(8)));

__global__ __launch_bounds__(256) void listmle_row_kernel(
    const float* __restrict__ logits,
    const long long* __restrict__ ids,
    const float* __restrict__ wts,
    float* __restrict__ row_loss)
{
    __shared__ float s_w[LL];     // weights (normalized, then sorted desc)
    __shared__ float s_x[LL];     // logits (gathered, then ordered)
    __shared__ float s_e[LL];     // exp(x - max)
    __shared__ float s_suf[LL];   // within-chunk suffix sums (WMMA output)
    __shared__ float s_tail[16];  // cross-chunk tail sums
    __shared__ float s_red[LL];   // reduction scratch

    const int b = blockIdx.x;
    const int t = threadIdx.x;

    // ---- gather ----
    const long long id = ids[(size_t)b * LL + t];          // int64 ids in [0, N)
    const float x = logits[(size_t)b * NN + (size_t)id];
    float w = wts[(size_t)b * LL + t];

    // ---- weight-sum reduction, normalize ----
    s_red[t] = w;
    __syncthreads();
    for (int s = 128; s > 0; s >>= 1) {
        if (t < s) s_red[t] += s_red[t + s];
        __syncthreads();
    }
    const float wsum = s_red[0];
    __syncthreads();
    w = w / fmaxf(wsum, EPSV);

    s_w[t] = w;
    s_x[t] = x;
    __syncthreads();

    // ---- bitonic sort of 256 (key = weight, descending; payload = logit) ----
    for (int k = 2; k <= LL; k <<= 1) {
        for (int j = k >> 1; j > 0; j >>= 1) {
            const int ixj = t ^ j;
            if (ixj > t) {
                const bool up = ((t & k) == 0);
                const float w0 = s_w[t], w1 = s_w[ixj];
                const bool do_swap = up ? (w0 < w1) : (w0 > w1); // descending network
                if (do_swap) {
                    const float x0 = s_x[t], x1 = s_x[ixj];
                    s_w[t] = w1; s_w[ixj] = w0;
                    s_x[t] = x1; s_x[ixj] = x0;
                }
            }
            __syncthreads();
        }
    }

    const float xo = s_x[t];
    const float wo = s_w[t];

    // ---- row max of ordered logits ----
    s_red[t] = xo;
    __syncthreads();
    for (int s = 128; s > 0; s >>= 1) {
        if (t < s) s_red[t] = fmaxf(s_red[t], s_red[t + s]);
        __syncthreads();
    }
    const float mx = s_red[0];
    __syncthreads();

    s_e[t] = expf(xo - mx);
    __syncthreads();

    // ---- suffix sums via WMMA: view e as E[16][16] (chunk r, pos c).
    // W = E x T, T[k][c] = (k >= c)  ->  W[r][c] = sum_{k>=c} E[r][k].
    // Done as 4 chained V_WMMA_F32_16X16X4_F32 by wave 0 (EXEC all-ones).
    if (t < 32) {
        const int lane  = t;
        const int mrow  = lane & 15;          // A: M = lane%16
        const int khalf = (lane >> 4) << 1;   // A/B: lanes>=16 hold K+2,K+3
        const int ncol  = lane & 15;          // B/D: N = lane%16
        v8f acc = {};
        #pragma unroll
        for (int kk = 0; kk < 16; kk += 4) {
            v2f a, bm;
            a.x = s_e[mrow * 16 + kk + khalf];
            a.y = s_e[mrow * 16 + kk + khalf + 1];
            const int kb = kk + khalf;
            bm.x = (kb     >= ncol) ? 1.0f : 0.0f;   // T[kb][n]
            bm.y = (kb + 1 >= ncol) ? 1.0f : 0.0f;   // T[kb+1][n]
            acc = __builtin_amdgcn_wmma_f32_16x16x4_f32(
                /*neg_a=*/false, a, /*neg_b=*/false, bm,
                /*c_mod=*/(short)0, acc, /*reuse_a=*/false, /*reuse_b=*/false);
        }
        const int mbase = (lane >> 4) << 3;   // D: M = v (+8 for upper half-wave)
        #pragma unroll
        for (int v = 0; v < 8; ++v) {
            s_suf[(v + mbase) * 16 + ncol] = acc[v];
        }
    }
    __syncthreads();

    // ---- cross-chunk tail: tail[r] = sum_{r' > r} rowtot[r'], rowtot = W[r][0] ----
    if (t < 16) {
        float s = 0.0f;
        for (int r = t + 1; r < 16; ++r) s += s_suf[r * 16];
        s_tail[t] = s;
    }
    __syncthreads();

    // ---- suffix lse, weighted loss, row reduction ----
    const float suf = s_suf[t] + s_tail[t >> 4];
    const float lse = mx + logf(suf);
    s_red[t] = wo * (lse - xo);
    __syncthreads();
    for (int s = 128; s > 0; s >>= 1) {
        if (t < s) s_red[t] += s_red[t + s];
        __syncthreads();
    }
    if (t == 0) row_loss[b] = s_red[0];
}

__global__ __launch_bounds__(256) void reduce_mean_kernel(
    const float* __restrict__ row_loss, float* __restrict__ out)
{
    __shared__ float s_red[256];
    const int t = threadIdx.x;
    float s = 0.0f;
    for (int i = t; i < BB; i += 256) s += row_loss[i];
    s_red[t] = s;
    __syncthreads();
    for (int k = 128; k > 0; k >>= 1) {
        if (t < k) s_red[t] += s_red[t + k];
        __syncthreads();
    }
    if (t == 0) out[0] = s_red[0] * (1.0f / (float)BB);
}

extern "C" void kernel_launch(void* const* d_in, const int* in_sizes, int n_in,
                              void* d_out, int out_size, void* d_ws, size_t ws_size,
                              hipStream_t stream) {
    const float*     logits = (const float*)d_in[0];
    const long long* ids    = (const long long*)d_in[1]; // int64
    const float*     wts    = (const float*)d_in[2];
    float* row_loss = (float*)d_ws;                      // 4096 * 4B = 16 KB scratch

    listmle_row_kernel<<<BB, 256, 0, stream>>>(logits, ids, wts, row_loss);
    reduce_mean_kernel<<<1, 256, 0, stream>>>(row_loss, (float*)d_out);
}